// HybridElmanCell_738734375258
// MI455X (gfx1250) — compile-verified
//
#include <hip/hip_runtime.h>
#include <math.h>
#include <stdint.h>

typedef __attribute__((ext_vector_type(2))) float v2f;
typedef __attribute__((ext_vector_type(8))) float v8f;

#define T_STEPS 256
#define B_DIM   32
#define C_DIM   1536
#define M_DIM   1536
#define NBLK_REC 48
#define EPSV    1e-8f
#define TARGETR 0.99f

// pregemm: block tile 64(M) x 128(N), BK=32, 8 waves (2x4), 32x32 per wave
#define PG_BM 64
#define PG_BN 128
#define PG_BK 32
#define PG_PAD 36      // gcd(36,64)=4 -> 16 fragment rows hit distinct banks; 8B-aligned rows
#define PG_NCH (C_DIM / PG_BK)
#define PG_CPT 6       // async copies per thread per chunk (2 for X + 4 for W)

// recurrent: chunk K=96, double buffered
#define RC_CHUNK 96
#define RC_PAD   100   // 100 % 64 == 36 -> distinct banks; 8B-aligned rows
#define RC_NCH (C_DIM / RC_CHUNK)
#define RC_CPT 12      // async copies per thread per chunk (6 for h + 6 for W)

__device__ __forceinline__ v8f wmma_f32(v2f a, v2f b, v8f c) {
  // D = A(16x4,f32) * B(4x16,f32) + C(16x16,f32)
  return __builtin_amdgcn_wmma_f32_16x16x4_f32(false, a, false, b, (short)0, c,
                                               false, false);
}

// gfx1250 async global->LDS copy, 16B per lane, tracked on ASYNCcnt.
// LDS byte address = low 32 bits of the generic pointer (LDS aperture rule).
__device__ __forceinline__ void async_g2l_b128(const float* gaddr, float* lds) {
  unsigned l = (unsigned)(uintptr_t)lds;
  asm volatile("global_load_async_to_lds_b128 %0, %1, off"
               :: "v"(l), "v"(gaddr) : "memory");
}
template <int N>
__device__ __forceinline__ void wait_async_le() {
  asm volatile("s_wait_asynccnt %0" :: "i"(N) : "memory");
}

__device__ __forceinline__ float sigmoidf_(float x) {
  return 1.0f / (1.0f + expf(-x));
}

// ---------------------------------------------------------------- init
__global__ void k_init(const float* __restrict__ h0_core, float* hbuf0,
                       float* u, int* cnt) {
  int i = blockIdx.x * 256 + threadIdx.x;            // grid covers B*C
  if (i < B_DIM * C_DIM) hbuf0[i] = h0_core[i];
  if (i < C_DIM) {                                   // deterministic pseudo-random u0
    float s = sinf((float)(i + 1) * 12.9898f) * 43758.5453f;
    u[i] = s - truncf(s);
  }
  if (i < T_STEPS) cnt[i] = 0;                       // re-zero barrier counters each call
}

// ------------------------------------------------------ vector normalize
__global__ void k_normalize(float* vec, int n, float* norm_out, float eps) {
  __shared__ float red[256];
  float s = 0.f;
  for (int i = threadIdx.x; i < n; i += 256) { float x = vec[i]; s += x * x; }
  red[threadIdx.x] = s;
  __syncthreads();
  for (int off = 128; off > 0; off >>= 1) {
    if (threadIdx.x < off) red[threadIdx.x] += red[threadIdx.x + off];
    __syncthreads();
  }
  float nrm = sqrtf(red[0]);
  if (threadIdx.x == 0) *norm_out = nrm;
  float inv = 1.0f / (nrm + eps);
  for (int i = threadIdx.x; i < n; i += 256) vec[i] *= inv;
}

// ------------------------------------------------------ v = W^T u  (column dot, coalesced)
__global__ void k_matvec_wt(const float* __restrict__ W,
                            const float* __restrict__ u, float* v) {
  __shared__ float us[C_DIM];
  for (int i = threadIdx.x; i < C_DIM; i += 256) us[i] = u[i];
  __syncthreads();
  int j = blockIdx.x * 256 + threadIdx.x;
  float acc = 0.f;
  for (int i = 0; i < C_DIM; ++i) acc += W[(size_t)i * C_DIM + j] * us[i];
  v[j] = acc;
}

// ------------------------------------------------------ u = W v  (wave per row)
__global__ void k_matvec_w(const float* __restrict__ W,
                           const float* __restrict__ v, float* u) {
  __shared__ float vs[C_DIM];
  for (int i = threadIdx.x; i < C_DIM; i += 256) vs[i] = v[i];
  __syncthreads();
  int wave = threadIdx.x >> 5, lane = threadIdx.x & 31;
  int row = blockIdx.x * 8 + wave;
  const float* Wr = W + (size_t)row * C_DIM;
  float acc = 0.f;
  for (int k = lane; k < C_DIM; k += 32) acc += Wr[k] * vs[k];
  for (int off = 16; off > 0; off >>= 1) acc += __shfl_down(acc, off, 32);
  if (lane == 0) u[row] = acc;
}

// ------------------------------------------------------ W_eff = W_h * 0.99/(sigma+eps)
__global__ void k_scale_w(const float* __restrict__ Wh, float* Weff,
                          const float* __restrict__ norm_last) {
  float nrm = *norm_last;                       // ||W v_final||
  float sigma = nrm * nrm / (nrm + EPSV);       // u.(Wv) with u = Wv/(||Wv||+eps)
  float sc = TARGETR / (sigma + EPSV);
  size_t i = (size_t)blockIdx.x * 256 + threadIdx.x;
  Weff[i] = Wh[i] * sc;
}

// ------------------------------------------------------ memory branch (streaming, non-temporal)
__global__ void k_mem(const float* __restrict__ xm, const float* __restrict__ zm,
                      const float* __restrict__ h0m, const float* __restrict__ a_mem,
                      float* __restrict__ out_m, float* __restrict__ hmem_f) {
  int idx = blockIdx.x * 256 + threadIdx.x;      // < B*M
  int m = idx % M_DIM;
  float d = sigmoidf_(a_mem[m]);
  float h = h0m[idx];
  const int BM = B_DIM * M_DIM;
  for (int t = 0; t < T_STEPS; ++t) {
    size_t o = (size_t)t * BM + idx;
    h = d * h + __builtin_nontemporal_load(&xm[o]);
    float z = __builtin_nontemporal_load(&zm[o]);
    __builtin_nontemporal_store(h * (z * sigmoidf_(z)), &out_m[o]);
  }
  hmem_f[idx] = h;
}

// ------------------------------------------------------ big GEMM: Y = X @ Wx^T + b
// X:[8192,1536]  Wx:[1536,1536]  Y -> out_core region (acts as Xp for the scan)
// 2x2 register blocking per wave + double-buffered async LDS staging.
__global__ void __launch_bounds__(256) k_pregemm(const float* __restrict__ X,
                                                 const float* __restrict__ Wx,
                                                 const float* __restrict__ bias,
                                                 float* __restrict__ Y) {
  __shared__ float Xs[2][PG_BM][PG_PAD];
  __shared__ float Ws[2][PG_BN][PG_PAD];
  const int tid = threadIdx.x;
  const int lane = tid & 31, wave = tid >> 5;
  const int lhi = lane >> 4, llo = lane & 15;
  const int waveM = wave >> 2, waveN = wave & 3;     // 2 x 4 wave grid, 32x32/wave
  const int mBase = blockIdx.x * PG_BM;
  const int nBase = blockIdx.y * PG_BN;

  v8f acc00 = {0.f,0.f,0.f,0.f,0.f,0.f,0.f,0.f};
  v8f acc01 = acc00, acc10 = acc00, acc11 = acc00;

  auto stage = [&](int kc, int buf) {
    int k0 = kc * PG_BK;
#pragma unroll
    for (int i = 0; i < 2; ++i) {                    // X tile: 64 x 32 = 512 f4
      int f = tid + i * 256;
      int row = f >> 3;
      int c4 = (f & 7) << 2;
      async_g2l_b128(X + (size_t)(mBase + row) * C_DIM + k0 + c4, &Xs[buf][row][c4]);
    }
#pragma unroll
    for (int i = 0; i < 4; ++i) {                    // W tile: 128 x 32 = 1024 f4
      int f = tid + i * 256;
      int row = f >> 3;
      int c4 = (f & 7) << 2;
      async_g2l_b128(Wx + (size_t)(nBase + row) * C_DIM + k0 + c4, &Ws[buf][row][c4]);
    }
  };

  stage(0, 0);
  for (int kc = 0; kc < PG_NCH; ++kc) {
    const int cur = kc & 1;
    if (kc + 1 < PG_NCH) {
      stage(kc + 1, cur ^ 1);                        // prefetch next chunk
      wait_async_le<PG_CPT>();                       // current chunk done (in-order retire)
    } else {
      wait_async_le<0>();
    }
    __syncthreads();
#pragma unroll
    for (int ks = 0; ks < PG_BK / 4; ++ks) {
      int koff = ks * 4 + 2 * lhi;                   // K = v + 2*(lane>=16)
      v2f a0 = *(const v2f*)&Xs[cur][waveM * 32 + llo][koff];
      v2f a1 = *(const v2f*)&Xs[cur][waveM * 32 + 16 + llo][koff];
      v2f b0 = *(const v2f*)&Ws[cur][waveN * 32 + llo][koff];
      v2f b1 = *(const v2f*)&Ws[cur][waveN * 32 + 16 + llo][koff];
      acc00 = wmma_f32(a0, b0, acc00);
      acc01 = wmma_f32(a0, b1, acc01);
      acc10 = wmma_f32(a1, b0, acc10);
      acc11 = wmma_f32(a1, b1, acc11);
    }
    __syncthreads();                                 // readers done before next overwrite
  }

  int col0 = nBase + waveN * 32 + llo;
  int col1 = col0 + 16;
  float bv0 = bias[col0], bv1 = bias[col1];
#pragma unroll
  for (int v = 0; v < 8; ++v) {                      // D: M = v + 8*(lane>=16)
    int r0 = mBase + waveM * 32 + v + 8 * lhi;
    int r1 = r0 + 16;
    Y[(size_t)r0 * C_DIM + col0] = acc00[v] + bv0;
    Y[(size_t)r0 * C_DIM + col1] = acc01[v] + bv1;
    Y[(size_t)r1 * C_DIM + col0] = acc10[v] + bv0;
    Y[(size_t)r1 * C_DIM + col1] = acc11[v] + bv1;
  }
}

// ------------------------------------------------------ persistent recurrent scan
// 48 blocks x 128 thr; block owns 32 output columns; device barrier per step.
// Double-buffered async LDS staging of h and the W slice.
__global__ void __launch_bounds__(128) k_recurrent(const float* __restrict__ Weff,
                                                   const float* __restrict__ z_core,
                                                   float* __restrict__ out_core, // Xp in, out in-place
                                                   float* __restrict__ hbuf,     // 2 * B*C
                                                   float* __restrict__ hcore_f,
                                                   int* __restrict__ cnt) {
  __shared__ float As[2][B_DIM][RC_PAD];
  __shared__ float Ws[2][32][RC_PAD];
  const int tid = threadIdx.x;
  const int lane = tid & 31, wave = tid >> 5;
  const int lhi = lane >> 4, llo = lane & 15;
  const int rowHalf = wave & 1, colHalf = wave >> 1; // 2x2 quadrants of 32x32 tile
  const int colbase = blockIdx.x * 32;

  for (int t = 0; t < T_STEPS; ++t) {
    const float* hcur = hbuf + (size_t)(t & 1) * (B_DIM * C_DIM);
    float* hnxt = hbuf + (size_t)((t + 1) & 1) * (B_DIM * C_DIM);
    v8f acc = {0.f,0.f,0.f,0.f,0.f,0.f,0.f,0.f};

    auto stage = [&](int kc, int buf) {
      int k0 = kc * RC_CHUNK;
#pragma unroll
      for (int i = 0; i < 6; ++i) {                  // 32 x 96 = 768 f4 each
        int f = tid + i * 128;
        int row = f / 24;
        int c4 = (f % 24) * 4;
        async_g2l_b128(hcur + (size_t)row * C_DIM + k0 + c4, &As[buf][row][c4]);
        async_g2l_b128(Weff + (size_t)(colbase + row) * C_DIM + k0 + c4,
                       &Ws[buf][row][c4]);
      }
    };

    stage(0, 0);
    for (int kc = 0; kc < RC_NCH; ++kc) {
      const int cur = kc & 1;
      if (kc + 1 < RC_NCH) {
        stage(kc + 1, cur ^ 1);
        wait_async_le<RC_CPT>();
      } else {
        wait_async_le<0>();
      }
      __syncthreads();
#pragma unroll
      for (int ks = 0; ks < RC_CHUNK / 4; ++ks) {
        int koff = ks * 4 + 2 * lhi;
        v2f bf = *(const v2f*)&Ws[cur][colHalf * 16 + llo][koff]; // B[k][n] = Weff[n][k]
        v2f af = *(const v2f*)&As[cur][rowHalf * 16 + llo][koff];
        acc = wmma_f32(af, bf, acc);
      }
      __syncthreads();
    }

    // epilogue: h = tanh(h@W^T + Xp); out = h * silu(z)
    size_t tb = (size_t)t * B_DIM * C_DIM;
    int col = colbase + colHalf * 16 + llo;
#pragma unroll
    for (int v = 0; v < 8; ++v) {
      int br = rowHalf * 16 + v + 8 * lhi;
      size_t o = tb + (size_t)br * C_DIM + col;
      float pre = acc[v] + out_core[o];              // Xp read, then overwritten below
      float h = tanhf(pre);
      float z = z_core[o];
      out_core[o] = h * (z * sigmoidf_(z));
      hnxt[(size_t)br * C_DIM + col] = h;
      if (t == T_STEPS - 1) hcore_f[(size_t)br * C_DIM + col] = h;
    }

    // device-wide barrier: all 48 blocks finish step t before t+1
    __threadfence();
    __syncthreads();
    if (tid == 0) {
      __hip_atomic_fetch_add(&cnt[t], 1, __ATOMIC_ACQ_REL, __HIP_MEMORY_SCOPE_AGENT);
      while (__hip_atomic_load(&cnt[t], __ATOMIC_ACQUIRE, __HIP_MEMORY_SCOPE_AGENT) < NBLK_REC)
        __builtin_amdgcn_s_sleep(1);
    }
    __syncthreads();
  }
}

// ==================================================================
extern "C" void kernel_launch(void* const* d_in, const int* in_sizes, int n_in,
                              void* d_out, int out_size, void* d_ws, size_t ws_size,
                              hipStream_t stream) {
  const float* x_core  = (const float*)d_in[0];
  const float* z_core  = (const float*)d_in[1];
  const float* x_mem   = (const float*)d_in[2];
  const float* z_mem   = (const float*)d_in[3];
  const float* h0_core = (const float*)d_in[4];
  const float* h0_mem  = (const float*)d_in[5];
  const float* W_x     = (const float*)d_in[6];
  const float* W_h     = (const float*)d_in[7];
  const float* b_core  = (const float*)d_in[8];
  const float* a_mem   = (const float*)d_in[9];

  float* out      = (float*)d_out;
  float* out_core = out;                                        // T*B*C
  float* out_mem  = out + (size_t)T_STEPS * B_DIM * C_DIM;      // T*B*M
  float* hcore_f  = out_mem + (size_t)T_STEPS * B_DIM * M_DIM;  // B*C
  float* hmem_f   = hcore_f + (size_t)B_DIM * C_DIM;            // B*M

  float* ws   = (float*)d_ws;
  size_t off  = 0;
  float* Weff = ws + off; off += (size_t)C_DIM * C_DIM;
  float* hbuf = ws + off; off += (size_t)2 * B_DIM * C_DIM;
  float* u    = ws + off; off += C_DIM;
  float* v    = ws + off; off += C_DIM;
  float* scal = ws + off; off += 8;
  int*   cnt  = (int*)(ws + off);

  // init: h0 -> hbuf[0], pseudo-random u0, zero barrier counters
  k_init<<<(B_DIM * C_DIM) / 256, 256, 0, stream>>>(h0_core, hbuf, u, cnt);
  k_normalize<<<1, 256, 0, stream>>>(u, C_DIM, scal + 0, 0.0f);

  // 3 power iterations; sigma comes from the last u-norm
  for (int it = 0; it < 3; ++it) {
    k_matvec_wt<<<C_DIM / 256, 256, 0, stream>>>(W_h, u, v);
    k_normalize<<<1, 256, 0, stream>>>(v, C_DIM, scal + 1, EPSV);
    k_matvec_w<<<C_DIM / 8, 256, 0, stream>>>(W_h, v, u);
    k_normalize<<<1, 256, 0, stream>>>(u, C_DIM, scal + 2, EPSV);
  }
  k_scale_w<<<(C_DIM * C_DIM) / 256, 256, 0, stream>>>(W_h, Weff, scal + 2);

  // hoisted input GEMM -> Xp stored in out_core region
  dim3 pgrid((T_STEPS * B_DIM) / PG_BM, C_DIM / PG_BN);
  k_pregemm<<<pgrid, 256, 0, stream>>>(x_core, W_x, b_core, out_core);

  // independent memory branch
  k_mem<<<(B_DIM * M_DIM) / 256, 256, 0, stream>>>(x_mem, z_mem, h0_mem, a_mem,
                                                   out_mem, hmem_f);

  // persistent recurrent scan
  k_recurrent<<<NBLK_REC, 128, 0, stream>>>(Weff, z_core, out_core, hbuf,
                                            hcore_f, cnt);
}